// DHCN_52913997086834
// MI455X (gfx1250) — compile-verified
//
#include <hip/hip_runtime.h>
#include <math.h>

// ---------------- problem constants ----------------
#define N_NODE 50000
#define EMB    100
#define BATCH  1024
#define SEQ    100
#define NNZ    800000
#define N_LAYERS 3
#define BETA   0.01f

typedef float v2f __attribute__((ext_vector_type(2)));
typedef float v8f __attribute__((ext_vector_type(8)));

// =====================================================================
// One-wave multi-tile WMMA f32 GEMM:
//   C[M x N] = act(A[M x K] @ W[K x N] + bias[n] + sessBias[(m/SEQ)*EMB + n])
// Each wave owns a 16-row x (16*NT)-col strip: the A fragment is loaded once
// per K-step and reused across NT independent accumulator chains (no WMMA
// RAW stalls between consecutive v_wmma on the same accumulator).
// ld(A)=K, ld(W)=ld(C)=Ncols.  M must be a multiple of 16.
// grid = (M/16, ceil(N/(16*NT))), block = 32 (one wave32).
// Fragment layouts per CDNA5 ISA 7.12.2 (32-bit A 16x4 / C-D 16x16).
// =====================================================================
template <int NT>
__global__ __launch_bounds__(32) void wmma_gemm_k(
    const float* __restrict__ A, const float* __restrict__ W,
    const float* __restrict__ bias, const float* __restrict__ sessBias,
    float* __restrict__ C, int K, int Ncols, int act)
{
    const int lane  = threadIdx.x;            // 0..31, EXEC all ones
    const int m0    = blockIdx.x * 16;
    const int n0    = blockIdx.y * (16 * NT);
    const int mrow  = m0 + (lane & 15);
    const int ncol  = lane & 15;
    const int khalf = (lane >> 4) << 1;       // 0 for lanes 0-15, 2 for 16-31

    int nn[NT];                               // clamped B column per tile
#pragma unroll
    for (int t = 0; t < NT; ++t) {
        const int n = n0 + t * 16 + ncol;
        nn[t] = (n < Ncols) ? n : (Ncols - 1);
    }

    v8f acc[NT];
#pragma unroll
    for (int t = 0; t < NT; ++t) acc[t] = (v8f){};

    const float* Arow = A + (size_t)mrow * K;
    for (int k0 = 0; k0 < K; k0 += 4) {
        const int kk = k0 + khalf;
        v2f a;
        a.x = Arow[kk];
        a.y = Arow[kk + 1];
        const float* W0 = W + (size_t)kk * Ncols;
        const float* W1 = W0 + Ncols;
#pragma unroll
        for (int t = 0; t < NT; ++t) {
            v2f b;
            b.x = W0[nn[t]];
            b.y = W1[nn[t]];
            acc[t] = __builtin_amdgcn_wmma_f32_16x16x4_f32(
                false, a, false, b, (short)0, acc[t], false, false);
        }
    }

    const int mbase = m0 + ((lane >> 4) << 3);   // rows r / r+8 split
#pragma unroll
    for (int t = 0; t < NT; ++t) {
        const int n = n0 + t * 16 + ncol;
        if (n < Ncols) {
#pragma unroll
            for (int r = 0; r < 8; ++r) {
                float v = acc[t][r];
                if (bias)     v += bias[n];
                if (sessBias) v += sessBias[(size_t)((mbase + r) / SEQ) * EMB + n];
                if (act == 1)      v = tanhf(v);
                else if (act == 2) v = 1.0f / (1.0f + expf(-v));
                C[(size_t)(mbase + r) * Ncols + n] = v;
            }
        }
    }
}

// GEMM1 with virtual A = concat([pos broadcast over batch], seq_h),
// full-width strip (NT=7 covers N=100): M=B*SEQ, K=2*EMB, N=EMB.
__global__ __launch_bounds__(32) void wmma_gemm_concat_k(
    const float* __restrict__ pos, const float* __restrict__ seqh,
    const float* __restrict__ w1w, const float* __restrict__ w1b,
    float* __restrict__ nh)
{
    constexpr int NT = 7;
    const int lane  = threadIdx.x;
    const int m0    = blockIdx.x * 16;
    const int mrow  = m0 + (lane & 15);
    const int lseq  = mrow % SEQ;
    const int ncol  = lane & 15;
    const int khalf = (lane >> 4) << 1;

    int nn[NT];
#pragma unroll
    for (int t = 0; t < NT; ++t) {
        const int n = t * 16 + ncol;
        nn[t] = (n < EMB) ? n : (EMB - 1);
    }

    v8f acc[NT];
#pragma unroll
    for (int t = 0; t < NT; ++t) acc[t] = (v8f){};

    const float* posrow = pos + (size_t)lseq * EMB;
    const float* seqrow = seqh + (size_t)mrow * EMB;
    for (int k0 = 0; k0 < 2 * EMB; k0 += 4) {
        const int kk = k0 + khalf;
        v2f a;
        a.x = (kk < EMB)     ? posrow[kk]     : seqrow[kk - EMB];
        a.y = (kk + 1 < EMB) ? posrow[kk + 1] : seqrow[kk + 1 - EMB];
        const float* W0 = w1w + (size_t)kk * EMB;
        const float* W1 = W0 + EMB;
#pragma unroll
        for (int t = 0; t < NT; ++t) {
            v2f b;
            b.x = W0[nn[t]];
            b.y = W1[nn[t]];
            acc[t] = __builtin_amdgcn_wmma_f32_16x16x4_f32(
                false, a, false, b, (short)0, acc[t], false, false);
        }
    }

    const int mbase = m0 + ((lane >> 4) << 3);
#pragma unroll
    for (int t = 0; t < NT; ++t) {
        const int n = t * 16 + ncol;
        if (n < EMB) {
#pragma unroll
            for (int r = 0; r < 8; ++r)
                nh[(size_t)(mbase + r) * EMB + n] = tanhf(acc[t][r] + w1b[n]);
        }
    }
}

// =====================================================================
// SpMM scatter: y[rows[e],:] += vals[e] * x[cols[e],:]   (one wave per edge)
// 20MB operand is L2-resident on MI455X (192MB L2) -> atomics stay in L2.
// =====================================================================
__global__ void spmm_scatter_k(const float* __restrict__ vals,
                               const int*   __restrict__ rows,
                               const int*   __restrict__ cols,
                               const float* __restrict__ x,
                               float*       __restrict__ y)
{
    const int wave = (blockIdx.x * blockDim.x + threadIdx.x) >> 5;
    const int lane = threadIdx.x & 31;
    if (wave >= NNZ) return;
    const float v = vals[wave];
    const float* xr = x + (size_t)cols[wave] * EMB;
    float*       yr = y + (size_t)rows[wave] * EMB;
    for (int j = lane; j < EMB; j += 32)
        unsafeAtomicAdd(&yr[j], v * xr[j]);
}

// ---------------- elementwise helpers ----------------
__global__ void fill_zero_k(float* p, int n)
{ int i = blockIdx.x * blockDim.x + threadIdx.x; if (i < n) p[i] = 0.0f; }

__global__ void copy2_k(const float* __restrict__ s, float* d1, float* d2, int n)
{ int i = blockIdx.x * blockDim.x + threadIdx.x; if (i < n) { float v = s[i]; d1[i] = v; d2[i] = v; } }

__global__ void axpy_k(float* __restrict__ y, const float* __restrict__ x, int n)
{ int i = blockIdx.x * blockDim.x + threadIdx.x; if (i < n) y[i] += x[i]; }

__global__ void scale_k(float* p, float a, int n)
{ int i = blockIdx.x * blockDim.x + threadIdx.x; if (i < n) p[i] *= a; }

// gather seq_h[b,l,:] = (idx==0) ? 0 : items[idx-1,:]   (one wave per (b,l))
__global__ void gather_seqh_k(const int* __restrict__ rev,
                              const float* __restrict__ items,
                              float* __restrict__ seqh)
{
    const int wave = (blockIdx.x * blockDim.x + threadIdx.x) >> 5;
    const int lane = threadIdx.x & 31;
    if (wave >= BATCH * SEQ) return;
    const int idx = rev[wave];
    float* dst = seqh + (size_t)wave * EMB;
    if (idx == 0) {
        for (int j = lane; j < EMB; j += 32) dst[j] = 0.0f;
    } else {
        const float* src = items + (size_t)(idx - 1) * EMB;
        for (int j = lane; j < EMB; j += 32) dst[j] = src[j];
    }
}

// hs[b,e] = sum_l seq_h[b,l,e] / slen[b]
__global__ void rowmean_k(const float* __restrict__ seqh,
                          const float* __restrict__ slen,
                          float* __restrict__ hs)
{
    int i = blockIdx.x * blockDim.x + threadIdx.x;
    if (i >= BATCH * EMB) return;
    const int b = i / EMB, e = i % EMB;
    const float* p = seqh + (size_t)b * SEQ * EMB + e;
    float sum = 0.0f;
    for (int l = 0; l < SEQ; ++l) sum += p[(size_t)l * EMB];
    hs[i] = sum / slen[b];
}

// s[b,e] = sum_l (idx ? emb[idx-1,e] : 0) / slen[b]
__global__ void gather_mean_k(const int* __restrict__ sess,
                              const float* __restrict__ emb,
                              const float* __restrict__ slen,
                              float* __restrict__ s)
{
    int i = blockIdx.x * blockDim.x + threadIdx.x;
    if (i >= BATCH * EMB) return;
    const int b = i / EMB, e = i % EMB;
    const int* row = sess + b * SEQ;
    float sum = 0.0f;
    for (int l = 0; l < SEQ; ++l) {
        const int idx = row[l];
        if (idx) sum += emb[(size_t)(idx - 1) * EMB + e];
    }
    s[i] = sum / slen[b];
}

// att[m] = (nh2[m,:] . w2) * mask[m]    (one wave per row)
__global__ void att_k(const float* __restrict__ nh2, const float* __restrict__ w2,
                      const int* __restrict__ mask, float* __restrict__ att)
{
    const int wave = (blockIdx.x * blockDim.x + threadIdx.x) >> 5;
    const int lane = threadIdx.x & 31;
    if (wave >= BATCH * SEQ) return;
    const float* row = nh2 + (size_t)wave * EMB;
    float p = 0.0f;
    for (int j = lane; j < EMB; j += 32) p += row[j] * w2[j];
    for (int off = 16; off > 0; off >>= 1) p += __shfl_down(p, off, 32);
    if (lane == 0) att[wave] = p * (float)mask[wave];
}

// sess_hgnn[b,e] = sum_l att[b,l] * seq_h[b,l,e]
__global__ void weighted_sum_k(const float* __restrict__ att,
                               const float* __restrict__ seqh,
                               float* __restrict__ out)
{
    int i = blockIdx.x * blockDim.x + threadIdx.x;
    if (i >= BATCH * EMB) return;
    const int b = i / EMB, e = i % EMB;
    const float* p = seqh + (size_t)b * SEQ * EMB + e;
    const float* a = att + b * SEQ;
    float sum = 0.0f;
    for (int l = 0; l < SEQ; ++l) sum += a[l] * p[(size_t)l * EMB];
    out[i] = sum;
}

// deterministic Fisher-Yates permutations (splitmix64), single thread
__global__ void perm_k(int* perm_r, int* perm_c)
{
    if (threadIdx.x != 0 || blockIdx.x != 0) return;
    unsigned long long st = 0x123456789ABCDEFull;
    auto next = [&]() -> unsigned long long {
        st += 0x9E3779B97F4A7C15ull;
        unsigned long long z = st;
        z = (z ^ (z >> 30)) * 0xBF58476D1CE4E5B9ull;
        z = (z ^ (z >> 27)) * 0x94D049BB133111EBull;
        return z ^ (z >> 31);
    };
    for (int i = 0; i < BATCH; ++i) perm_r[i] = i;
    for (int i = BATCH - 1; i > 0; --i) {
        int j = (int)(next() % (unsigned long long)(i + 1));
        int t = perm_r[i]; perm_r[i] = perm_r[j]; perm_r[j] = t;
    }
    for (int i = 0; i < EMB; ++i) perm_c[i] = i;
    for (int i = EMB - 1; i > 0; --i) {
        int j = (int)(next() % (unsigned long long)(i + 1));
        int t = perm_c[i]; perm_c[i] = perm_c[j]; perm_c[j] = t;
    }
}

// contrastive loss, fully deterministic single-block tree reduction
__global__ __launch_bounds__(1024) void contrast_k(
    const float* __restrict__ hg, const float* __restrict__ lg,
    const int* __restrict__ perm_r, const int* __restrict__ perm_c,
    float* __restrict__ out_scalar)
{
    __shared__ float red[1024];
    const int b = threadIdx.x;
    const float* hb = hg + (size_t)b * EMB;
    const float* lb = lg + (size_t)b * EMB;
    const float* cb = hg + (size_t)perm_r[b] * EMB;
    float pos = 0.0f, neg = 0.0f;
    for (int e = 0; e < EMB; ++e) {
        pos += hb[e] * lb[e];
        neg += lb[e] * cb[perm_c[e]];
    }
    const float sp = 1.0f / (1.0f + expf(-pos));
    const float sn = 1.0f / (1.0f + expf(-neg));
    red[b] = -logf(1e-8f + sp) - logf(1e-8f + 1.0f - sn);
    __syncthreads();
    for (int stride = 512; stride > 0; stride >>= 1) {
        if (b < stride) red[b] += red[b + stride];
        __syncthreads();
    }
    if (b == 0) *out_scalar = BETA * red[0];
}

// =====================================================================
extern "C" void kernel_launch(void* const* d_in, const int* in_sizes, int n_in,
                              void* d_out, int out_size, void* d_ws, size_t ws_size,
                              hipStream_t stream)
{
    const float* emb    = (const float*)d_in[0];
    const float* pos    = (const float*)d_in[1];
    const float* w1w    = (const float*)d_in[2];
    const float* w1b    = (const float*)d_in[3];
    const float* w2     = (const float*)d_in[4];
    const float* glu1w  = (const float*)d_in[5];
    const float* glu1b  = (const float*)d_in[6];
    const float* glu2w  = (const float*)d_in[7];
    const float* avals  = (const float*)d_in[8];
    const int*   arows  = (const int*)d_in[9];
    const int*   acols  = (const int*)d_in[10];
    const int*   sitem  = (const int*)d_in[11];
    const float* slen   = (const float*)d_in[12];
    const float* Dm     = (const float*)d_in[13];
    const float* Am     = (const float*)d_in[14];
    const int*   rsitem = (const int*)d_in[15];
    const int*   maskp  = (const int*)d_in[16];

    float* out   = (float*)d_out;
    float* items = out;                 // N_NODE*EMB = 5,000,000
    float* hgnn  = out + 5000000;       // BATCH*EMB  = 102,400
    float* scal  = out + 5102400;       // 1

    // ---------- workspace layout (floats) ----------
    float* ws    = (float*)d_ws;
    float* curA  = ws;                  // 5,000,000
    float* curB  = curA  + 5000000;     // 5,000,000
    float* seqh  = curB  + 5000000;     // 10,240,000
    float* nh    = seqh  + 10240000;    // 10,240,000
    float* nh2   = nh    + 10240000;    // 10,240,000
    float* hs    = nh2   + 10240000;    // 102,400
    float* hsg   = hs    + 102400;      // 102,400
    float* att   = hsg   + 102400;      // 102,400
    float* sbuf  = att   + 102400;      // 102,400
    float* DA    = sbuf  + 102400;      // 1,048,576
    float* lgA   = DA    + 1048576;     // 102,400
    float* lgB   = lgA   + 102400;      // 102,400
    float* lgAcc = lgB   + 102400;      // 102,400
    int*   permR = (int*)(lgAcc + 102400);
    int*   permC = permR + BATCH;

    const int NE  = N_NODE * EMB;       // 5,000,000
    const int BE  = BATCH * EMB;        // 102,400
    const int BL  = BATCH * SEQ;        // 102,400
    const int TPB = 256;

    // ---- 1) hypergraph propagation: items = (emb + sum_i spmm^i(emb)) / 4 ----
    copy2_k<<<(NE + TPB - 1) / TPB, TPB, 0, stream>>>(emb, curA, items, NE);
    {
        float* pa = curA; float* pb = curB;
        const int spmmBlocks = (NNZ * 32 + TPB - 1) / TPB;
        for (int l = 0; l < N_LAYERS; ++l) {
            fill_zero_k<<<(NE + TPB - 1) / TPB, TPB, 0, stream>>>(pb, NE);
            spmm_scatter_k<<<spmmBlocks, TPB, 0, stream>>>(avals, arows, acols, pa, pb);
            axpy_k<<<(NE + TPB - 1) / TPB, TPB, 0, stream>>>(items, pb, NE);
            float* t = pa; pa = pb; pb = t;
        }
    }
    scale_k<<<(NE + TPB - 1) / TPB, TPB, 0, stream>>>(items, 1.0f / (N_LAYERS + 1), NE);

    // ---- 2) seq_h gather + hs ----
    gather_seqh_k<<<(BL * 32 + TPB - 1) / TPB, TPB, 0, stream>>>(rsitem, items, seqh);
    rowmean_k<<<(BE + TPB - 1) / TPB, TPB, 0, stream>>>(seqh, slen, hs);

    // ---- 3) nh = tanh([pos || seq_h] @ w1_w + w1_b)  (WMMA f32, NT=7 strip) ----
    wmma_gemm_concat_k<<<dim3(BL / 16, 1), 32, 0, stream>>>(pos, seqh, w1w, w1b, nh);

    // ---- 4) hsg = hs @ glu2_w ; nh2 = sigmoid(nh @ glu1_w + glu1_b + hsg) ----
    wmma_gemm_k<2><<<dim3(BATCH / 16, 4), 32, 0, stream>>>(
        hs, glu2w, nullptr, nullptr, hsg, EMB, EMB, 0);
    wmma_gemm_k<7><<<dim3(BL / 16, 1), 32, 0, stream>>>(
        nh, glu1w, glu1b, hsg, nh2, EMB, EMB, 2);

    // ---- 5) attention + weighted sum -> sess_hgnn ----
    att_k<<<(BL * 32 + TPB - 1) / TPB, TPB, 0, stream>>>(nh2, w2, maskp, att);
    weighted_sum_k<<<(BE + TPB - 1) / TPB, TPB, 0, stream>>>(att, seqh, hgnn);

    // ---- 6) line-graph branch: s, DA = D@A, 3x propagation ----
    gather_mean_k<<<(BE + TPB - 1) / TPB, TPB, 0, stream>>>(sitem, emb, slen, sbuf);
    wmma_gemm_k<4><<<dim3(BATCH / 16, BATCH / 64), 32, 0, stream>>>(
        Dm, Am, nullptr, nullptr, DA, BATCH, BATCH, 0);
    copy2_k<<<(BE + TPB - 1) / TPB, TPB, 0, stream>>>(sbuf, lgA, lgAcc, BE);
    {
        float* pa = lgA; float* pb = lgB;
        for (int l = 0; l < N_LAYERS; ++l) {
            wmma_gemm_k<2><<<dim3(BATCH / 16, 4), 32, 0, stream>>>(
                DA, pa, nullptr, nullptr, pb, BATCH, EMB, 0);
            axpy_k<<<(BE + TPB - 1) / TPB, TPB, 0, stream>>>(lgAcc, pb, BE);
            float* t = pa; pa = pb; pb = t;
        }
    }
    scale_k<<<(BE + TPB - 1) / TPB, TPB, 0, stream>>>(lgAcc, 1.0f / (N_LAYERS + 1), BE);

    // ---- 7) contrastive scalar (deterministic) ----
    perm_k<<<1, 1, 0, stream>>>(permR, permC);
    contrast_k<<<1, 1024, 0, stream>>>(hgnn, lgAcc, permR, permC, scal);
}